// MultiThresholdLIFSpike_79826262163569
// MI455X (gfx1250) — compile-verified
//
#include <hip/hip_runtime.h>

// ---------------------------------------------------------------------------
// MultiThresholdLIFSpike, MI455X (gfx1250)
//
// x:   [T=16, B=32, C=128, H=32, W=32] fp32   (67,108,864 elements, 256 MiB)
// out: same shape. Recurrence along T only -> one register-resident chain per
// spatial element. Memory-bound: 512 MiB total traffic / 23.3 TB/s ~= 23 us.
//
// Strategy: float4 (b128) streaming with non-temporal hints (working set >>
// 192 MB L2, zero reuse), register double-buffering of the next time plane,
// and ungated gfx1250 global_prefetch_b8 on the plane after that (same-line
// requests coalesce in the VMEM address unit; gating per-cacheline costs more
// in EXEC save/restore SALU than it saves).
// Sigmoid via hardware v_exp_f32 + v_rcp_f32; beta*log2e*thr_k folded into
// wave-uniform constants.
//
// Trans budget: 32 trans wave-instrs per wave per plane -> ~21 B/clk/SIMD
// sustainable, vs ~9 B/clk/SIMD needed to saturate 23.3 TB/s: HBM-bound.
// No contraction anywhere in the op -> WMMA cannot reduce the traffic floor.
// ---------------------------------------------------------------------------

typedef float v4f __attribute__((ext_vector_type(4)));

#define LIF_T      16
#define LIF_TAU    0.25f
#define LIF_BETA   10.0f
#define LIF_L2E    1.4426950408889634f   // log2(e)

// sigmoid(beta*(u - thr_k)) = 1 / (1 + exp2(d_k - bl))
//   d_k = beta*log2e*thr_k  (uniform),  bl = beta*log2e*u
__device__ __forceinline__ float lif_spike(float u, float d0, float d1,
                                           float d2, float d3) {
    const float bl = u * (LIF_BETA * LIF_L2E);
    float s;
    s  = __builtin_amdgcn_rcpf(1.0f + __builtin_amdgcn_exp2f(d0 - bl));
    s += __builtin_amdgcn_rcpf(1.0f + __builtin_amdgcn_exp2f(d1 - bl));
    s += __builtin_amdgcn_rcpf(1.0f + __builtin_amdgcn_exp2f(d2 - bl));
    s += __builtin_amdgcn_rcpf(1.0f + __builtin_amdgcn_exp2f(d3 - bl));
    return s;
}

__device__ __forceinline__ float lif_sigmoid(float z) {
    return __builtin_amdgcn_rcpf(1.0f + __builtin_amdgcn_exp2f(-z * LIF_L2E));
}

__global__ __launch_bounds__(256)
void MultiThresholdLIFSpike_79826262163569_kernel(
        const float* __restrict__ x,
        const float* __restrict__ thr_raw,
        float* __restrict__ out,
        int plane4 /* elements-per-time-plane / 4 */) {

    const int i4 = blockIdx.x * 256 + threadIdx.x;
    if (i4 >= plane4) return;

    // Uniform thresholds (4 floats; wave-uniform address -> scalar loads).
    // thresholds = V_MIN + (V_MAX-V_MIN)*sigmoid(raw) = sigmoid(raw).
    const float d0 = lif_sigmoid(thr_raw[0]) * (LIF_BETA * LIF_L2E);
    const float d1 = lif_sigmoid(thr_raw[1]) * (LIF_BETA * LIF_L2E);
    const float d2 = lif_sigmoid(thr_raw[2]) * (LIF_BETA * LIF_L2E);
    const float d3 = lif_sigmoid(thr_raw[3]) * (LIF_BETA * LIF_L2E);

    const v4f* __restrict__ xv = (const v4f*)x;
    v4f*       __restrict__ ov = (v4f*)out;

    v4f mem; mem.x = mem.y = mem.z = mem.w = 0.0f;
    v4f spk = mem;

    // Prime the pipeline: plane t=0 in registers.
    v4f cur = __builtin_nontemporal_load(xv + i4);

#pragma unroll
    for (int t = 0; t < LIF_T; ++t) {
        // Keep L2 two planes ahead of the stream (gfx1250 global_prefetch_b8).
        if (t + 2 < LIF_T)
            __builtin_prefetch(
                (const void*)(xv + (size_t)(t + 2) * plane4 + i4), 0, 1);

        // Double-buffer the next plane in registers while computing this one.
        v4f nxt;
        if (t + 1 < LIF_T)
            nxt = __builtin_nontemporal_load(xv + (size_t)(t + 1) * plane4 + i4);
        else
            nxt = cur;  // dead value

        // u = tau * mem * (1 - spk) + x_t   (elementwise on 4 lanes)
        v4f u = LIF_TAU * mem * (1.0f - spk) + cur;

        v4f s;
        s.x = lif_spike(u.x, d0, d1, d2, d3);
        s.y = lif_spike(u.y, d0, d1, d2, d3);
        s.z = lif_spike(u.z, d0, d1, d2, d3);
        s.w = lif_spike(u.w, d0, d1, d2, d3);

        __builtin_nontemporal_store(s, ov + (size_t)t * plane4 + i4);

        mem = u;
        spk = s;
        cur = nxt;
    }
}

extern "C" void kernel_launch(void* const* d_in, const int* in_sizes, int n_in,
                              void* d_out, int out_size, void* d_ws, size_t ws_size,
                              hipStream_t stream) {
    const float* x       = (const float*)d_in[0];   // [T,B,C,H,W] fp32
    const float* thr_raw = (const float*)d_in[1];   // [4] fp32
    float*       out     = (float*)d_out;

    const int total  = in_sizes[0];          // 67,108,864
    const int plane  = total / LIF_T;        // 4,194,304 (divisible by 4)
    const int plane4 = plane / 4;            // 1,048,576 float4 chains

    const int block = 256;                   // 8 wave32s
    const int grid  = (plane4 + block - 1) / block;

    MultiThresholdLIFSpike_79826262163569_kernel<<<grid, block, 0, stream>>>(
        x, thr_raw, out, plane4);
}